// Top_Left_Corner_66623532695949
// MI455X (gfx1250) — compile-verified
//
#include <hip/hip_runtime.h>

// Problem constants (fixed by the reference: x is (16, 256, 128, 128) f32, NCHW)
#define HH 128
#define WW 128
#define PLANE (HH * WW)          // 16384 floats = 64 KB per plane
#define BLOCK 128                // 4 wave32s per block, one plane per block

typedef float float4v __attribute__((ext_vector_type(4)));
typedef int   int4v   __attribute__((ext_vector_type(4)));

#if defined(__HIP_DEVICE_COMPILE__)
// Pointer types matching this toolchain's async-load builtin signature:
//   void __builtin_amdgcn_global_load_async_to_lds_b128(int4 AS1*, int4 AS3*, imm, imm)
typedef __attribute__((address_space(1))) int4v* glob_b128_t;
typedef __attribute__((address_space(3))) int4v* lds_b128_t;
#endif

// out = 2 * revcummax_W( revcummax_H( x ) ), per (n,c) plane.
__global__ __launch_bounds__(BLOCK) void corner_pool_kernel(
    const float* __restrict__ x, float* __restrict__ out, int nplanes) {

  __shared__ float lds[PLANE];   // exactly 64 KB; 5 blocks fit per 320 KB WGP

  const int p = blockIdx.x;
  if (p >= nplanes) return;
  const int t = threadIdx.x;                 // 0..127

  const float* __restrict__ xp = x + (size_t)p * PLANE;
  float* __restrict__ op       = out + (size_t)p * PLANE;

  // ------------------------------------------------------------------
  // Phase 0: stage the 64 KB plane HBM -> LDS.
  // Preferred path: gfx1250 async direct-to-LDS B128 loads (ASYNCcnt),
  // bypassing VGPRs. 32 iterations x 128 lanes x 16 B = 64 KB, coalesced.
  // ------------------------------------------------------------------
#if defined(__HIP_DEVICE_COMPILE__) && \
    __has_builtin(__builtin_amdgcn_global_load_async_to_lds_b128)
  {
#pragma unroll
    for (int it = 0; it < PLANE / (4 * BLOCK); ++it) {   // 32 iters
      const int v = it * BLOCK + t;                      // float4 index
      __builtin_amdgcn_global_load_async_to_lds_b128(
          (glob_b128_t)(xp + (size_t)v * 4),
          (lds_b128_t)(&lds[v * 4]),
          /*offset=*/0, /*cpol=*/0);
    }
#if __has_builtin(__builtin_amdgcn_s_wait_asynccnt)
    __builtin_amdgcn_s_wait_asynccnt(0);
#else
    asm volatile("s_wait_asynccnt 0" ::: "memory");
#endif
  }
#else
  {
#pragma unroll
    for (int it = 0; it < PLANE / (4 * BLOCK); ++it) {
      const int v = it * BLOCK + t;
      *(float4v*)&lds[v * 4] = *(const float4v*)(xp + (size_t)v * 4);
    }
  }
#endif
  __syncthreads();

  // ------------------------------------------------------------------
  // Phase 1: reverse cummax along H (top pooling), in LDS.
  // Thread t owns column t. For a fixed h, lanes hit addresses
  // h*128 + lane -> 32 distinct banks: conflict-free.
  // Loads are address-independent of the previous store, so the
  // compiler can prefetch ahead of the max-chain.
  // ------------------------------------------------------------------
  {
    float v = lds[(HH - 1) * WW + t];
#pragma unroll 4
    for (int h = HH - 2; h >= 0; --h) {
      const float u = lds[h * WW + t];
      v = fmaxf(u, v);
      lds[h * WW + t] = v;
    }
  }
  __syncthreads();

  // ------------------------------------------------------------------
  // Phase 2: reverse cummax along W (left pooling), one wave32 per row.
  // Each 32-wide chunk does a 5-step shfl_down suffix-max; a broadcast
  // carry (lane 0 holds the chunk's running max) links the 4 chunks
  // from w=127 down to w=0.  LDS reads: addr = r*128 + (c*32 + lane)
  // -> 32 distinct banks, conflict-free, no padding needed.
  // The x2 and the global store are fused here (coalesced 128 B per
  // chunk-store), so results never go back through LDS.
  // ------------------------------------------------------------------
  {
    const int wave = t >> 5;        // 0..3
    const int lane = t & 31;

    for (int r = wave; r < HH; r += BLOCK / 32) {   // 32 rows per wave
      float carry = -__builtin_inff();
#pragma unroll
      for (int c = (WW / 32) - 1; c >= 0; --c) {    // chunks: w high -> low
        const int w = c * 32 + lane;
        float v = lds[r * WW + w];
        // intra-wave suffix max (lanes >= self); OOB shfl_down returns
        // own value, which is the identity under max.
        v = fmaxf(v, __shfl_down(v, 1));
        v = fmaxf(v, __shfl_down(v, 2));
        v = fmaxf(v, __shfl_down(v, 4));
        v = fmaxf(v, __shfl_down(v, 8));
        v = fmaxf(v, __shfl_down(v, 16));
        v = fmaxf(v, carry);                        // fold in higher chunks
        op[r * WW + w] = v + v;                     // fused "y + y"
        carry = __shfl(v, 0);                       // max over w >= c*32
      }
    }
  }
}

extern "C" void kernel_launch(void* const* d_in, const int* in_sizes, int n_in,
                              void* d_out, int out_size, void* d_ws, size_t ws_size,
                              hipStream_t stream) {
  (void)n_in; (void)d_ws; (void)ws_size; (void)out_size;
  const float* x = (const float*)d_in[0];
  float* out = (float*)d_out;
  const int nplanes = in_sizes[0] / PLANE;          // 16*256 = 4096
  corner_pool_kernel<<<dim3(nplanes), dim3(BLOCK), 0, stream>>>(x, out, nplanes);
}